// Middle_Moudle_v3_58815282152232
// MI455X (gfx1250) — compile-verified
//
#include <hip/hip_runtime.h>
#include <math.h>
#include <stdint.h>

// CDNA5 WMMA vector types
typedef __attribute__((ext_vector_type(2))) float v2f;
typedef __attribute__((ext_vector_type(8))) float v8f;

#define B_   8
#define S_   75
#define C_   64
#define P_   361      // W*W = 19*19 spatial positions
#define PT   23       // ceil(361/16) tiles along M and N
#define NPAIR (B_ * S_)
#define EPSV 1e-8f

// LDS layout in floats:
//  [0, 23104)        query block  [64][361]  (contiguous, same as global layout)
//  [23104, 23120)    16 zero-pad floats (last N-tile B-fragment reads land here)
//  [23120, 23488)    qnorm[368]
//  [23488, 23856)    snorm[368]
#define QN_OFF 23120
#define SN_OFF 23488
#define LDS_FLOATS 23856

__global__ __launch_bounds__(256)
void cosmax_wmma_kernel(const float* __restrict__ support,
                        const float* __restrict__ query,
                        float* __restrict__ out)
{
    extern __shared__ float lds[];
    const int pair = blockIdx.x;                       // b*S + s
    const float* __restrict__ qbase = query   + (size_t)pair * (C_ * P_);
    const float* __restrict__ sbase = support + (size_t)pair * (C_ * P_);
    const int tid = threadIdx.x;

    // ---- 1) async copy: query[64][361] -> LDS, 16B per lane per issue ----
    // GLOBAL_LOAD_ASYNC_TO_LDS_B128: global -> LDS without VGPR staging (ASYNCcnt).
    {
        const int n16 = (C_ * P_) / 4;                 // 5776 x 16B chunks
        for (int i = tid; i < n16; i += 256) {
            uint64_t gaddr = (uint64_t)(uintptr_t)(qbase + 4 * i);
            uint32_t laddr = (uint32_t)(uintptr_t)(lds + 4 * i);   // LDS byte offset (AS3 ptr)
            asm volatile("global_load_async_to_lds_b128 %0, %1, off"
                         :: "v"(laddr), "v"(gaddr) : "memory");
        }
        if (tid < 16) lds[C_ * P_ + tid] = 0.0f;       // zero pad tail
        asm volatile("s_wait_asynccnt 0" ::: "memory");
    }
    __syncthreads();

    // ---- 2) norms: qnorm from LDS, snorm from global ----
    for (int n = tid; n < 368; n += 256) {
        if (n < P_) {
            float aq = 0.f, as = 0.f;
            for (int c = 0; c < C_; ++c) {
                float vq = lds[c * P_ + n];   aq = fmaf(vq, vq, aq);
                float vs = sbase[c * P_ + n]; as = fmaf(vs, vs, as);
            }
            lds[QN_OFF + n] = sqrtf(aq);
            lds[SN_OFF + n] = sqrtf(as);
        } else {
            lds[QN_OFF + n] = 1.0f;                    // benign values for padded cols
            lds[SN_OFF + n] = 1.0f;
        }
    }
    __syncthreads();

    const int lane = tid & 31;
    const int wave = tid >> 5;
    const int half = lane >> 4;      // 0: lanes 0-15, 1: lanes 16-31
    const int l16  = lane & 15;
    const int koff = half * 2;       // upper half-wave holds K+2 / K+3 planes

    // ---- 3) tiled GEMM + masked row-max, one 16-row M-tile per wave iteration ----
    for (int mt = wave; mt < PT; mt += 8) {
        const int m0 = mt * 16;

        // Preload A fragments for all 16 K-steps (16x4 fp32 A layout):
        //   VGPR0: K=k0+koff, VGPR1: K=k0+1+koff; lane l16 = row m0+l16
        v2f afrag[16];
        {
            int mm = m0 + l16; if (mm > P_ - 1) mm = P_ - 1;   // clamp tail rows
            const float* __restrict__ pa = sbase + koff * P_ + mm;
#pragma unroll
            for (int kk = 0; kk < 16; ++kk) {
                const float* p = pa + (4 * kk) * P_;
                afrag[kk].x = p[0];
                afrag[kk].y = p[P_];
            }
        }

        // Support norms for this lane's 8 output rows (row = m0 + r + 8*half)
        float sn[8];
#pragma unroll
        for (int r = 0; r < 8; ++r) {
            int row = m0 + r + 8 * half; if (row > P_ - 1) row = P_ - 1;
            sn[r] = lds[SN_OFF + row];
        }

        float rowmax[8];
#pragma unroll
        for (int r = 0; r < 8; ++r) rowmax[r] = -INFINITY;

        for (int nt = 0; nt < PT; ++nt) {
            const int n = nt * 16 + l16;               // this lane's N column
            v8f acc = {};
#pragma unroll
            for (int kk = 0; kk < 16; ++kk) {
                // B fragment (4x16 fp32): VGPR0: K=k0+koff, VGPR1: K=k0+1+koff, col = n
                const float* __restrict__ q = lds + (4 * kk + koff) * P_ + n;
                v2f bfrag;
                bfrag.x = q[0];
                bfrag.y = q[P_];
                acc = __builtin_amdgcn_wmma_f32_16x16x4_f32(
                        /*neg_a=*/false, afrag[kk],
                        /*neg_b=*/false, bfrag,
                        /*c_mod=*/(short)0, acc,
                        /*reuse_a=*/false, /*reuse_b=*/false);
            }
            const float qn = lds[QN_OFF + n];
            const bool nvalid = (n < P_);
#pragma unroll
            for (int r = 0; r < 8; ++r) {
                // v_rcp_f32 (~1 ulp) instead of full IEEE div sequence
                float denom = fmaxf(sn[r] * qn, EPSV);
                float sim = acc[r] * __builtin_amdgcn_rcpf(denom);
                sim = nvalid ? sim : -INFINITY;
                rowmax[r] = fmaxf(rowmax[r], sim);
            }
        }

        // Cross-lane max within each 16-lane half (wave32 shuffles)
#pragma unroll
        for (int r = 0; r < 8; ++r) {
            float v = rowmax[r];
            v = fmaxf(v, __shfl_xor(v, 1, 32));
            v = fmaxf(v, __shfl_xor(v, 2, 32));
            v = fmaxf(v, __shfl_xor(v, 4, 32));
            v = fmaxf(v, __shfl_xor(v, 8, 32));
            rowmax[r] = v;
        }

        if (l16 == 0) {
#pragma unroll
            for (int r = 0; r < 8; ++r) {
                int row = m0 + r + 8 * half;
                if (row < P_) out[(size_t)pair * P_ + row] = rowmax[r];
            }
        }
    }
}

extern "C" void kernel_launch(void* const* d_in, const int* in_sizes, int n_in,
                              void* d_out, int out_size, void* d_ws, size_t ws_size,
                              hipStream_t stream) {
    (void)in_sizes; (void)n_in; (void)d_ws; (void)ws_size; (void)out_size;
    const float* support = (const float*)d_in[0];
    const float* query   = (const float*)d_in[1];
    float* out = (float*)d_out;

    const size_t shmem = (size_t)LDS_FLOATS * sizeof(float);   // ~95 KB (CDNA5: 320 KB/WGP)
    // Allow >64KB dynamic LDS (idempotent; not a stream op, graph-capture safe)
    (void)hipFuncSetAttribute((const void*)cosmax_wmma_kernel,
                              hipFuncAttributeMaxDynamicSharedMemorySize,
                              (int)shmem);

    cosmax_wmma_kernel<<<NPAIR, 256, shmem, stream>>>(support, query, out);
}